// BilinearInterpolation_6889127543174
// MI455X (gfx1250) — compile-verified
//
#include <hip/hip_runtime.h>

typedef __attribute__((ext_vector_type(2))) float v2f;
typedef __attribute__((ext_vector_type(4))) float v4f;
typedef __attribute__((ext_vector_type(8))) float v8f;

namespace {
constexpr int kOutH = 224, kOutW = 224;
constexpr int kImgH = 256, kImgW = 256;
constexpr int kC = 16;
constexpr int kBatch = 32;
constexpr int kPixPerWave = 16;    // one WMMA coordinate tile
constexpr int kWavesPerBlock = 8;  // 256 threads (8 wave32)
}

__global__ __launch_bounds__(256)
void st_bilinear_wmma(const float* __restrict__ img,
                      const float* __restrict__ theta,
                      float* __restrict__ out)
{
    const int lane = threadIdx.x & 31;
    const int wave = threadIdx.x >> 5;
    const int tile = blockIdx.x * kWavesPerBlock + wave;
    const int pbase = tile * kPixPerWave;

    const int b   = pbase / (kOutH * kOutW);
    const int rem = pbase - b * (kOutH * kOutW);
    const int oy  = rem / kOutW;
    const int oxb = rem - oy * kOutW;        // multiple of 16; 16 | 224 -> in-row tile

    const int n  = lane & 15;                // pixel in tile == WMMA row/col index
    const int hi = lane >> 4;                // channel half: 0 -> c0..7, 1 -> c8..15
    const int ox = oxb + n;

    // regular grid in [-1, 1] (linspace(-1, 1, 224))
    const float step = 2.0f / 223.0f;
    const float gx = fmaf((float)ox, step, -1.0f);
    const float gy = fmaf((float)oy, step, -1.0f);

    // theta row-major [2,3] for this batch
    const float* th = theta + b * 6;
    const float t00 = th[0], t01 = th[1], t02 = th[2];
    const float t10 = th[3], t11 = th[4], t12 = th[5];

    // ---- einsum('bij,jp->bip') on the WMMA unit: D = A(16x4) x B(4x16) ----
    // A: rows 0,1 = theta rows (K=0..2 used, K=3 = 0), rows 2..15 = 0.
    //    ISA layout: lane holds M = lane%16; v[0] = K0 (lanes 0-15) / K2 (16-31),
    //                v[1] = K1 / K3.
    // B: rows = [gx; gy; 1; 0], column N = lane%16, same K split across halves.
    const float a0 = (n == 0) ? (hi ? t02 : t00)
                   : (n == 1) ? (hi ? t12 : t10) : 0.0f;
    const float a1 = (n == 0) ? (hi ? 0.0f : t01)
                   : (n == 1) ? (hi ? 0.0f : t11) : 0.0f;
    const float b0 = hi ? 1.0f : gx;
    const float b1 = hi ? 0.0f : gy;

    v2f Av; Av[0] = a0; Av[1] = a1;
    v2f Bv; Bv[0] = b0; Bv[1] = b1;
    v8f Cv = {};
    Cv = __builtin_amdgcn_wmma_f32_16x16x4_f32(false, Av, false, Bv,
                                               (short)0, Cv, false, false);

    // D layout: VGPR0 = row M=0 (x), VGPR1 = row M=1 (y), N = lane (lanes 0-15).
    // Broadcast pixel n's coords to both channel-half lanes via ds_bpermute.
    const float xs = __int_as_float(
        __builtin_amdgcn_ds_bpermute(n << 2, __float_as_int(Cv[0])));
    const float ys = __int_as_float(
        __builtin_amdgcn_ds_bpermute(n << 2, __float_as_int(Cv[1])));

    // [-1,1] -> pixel coordinates
    const float px = 0.5f * (xs + 1.0f) * (float)kImgW;
    const float py = 0.5f * (ys + 1.0f) * (float)kImgH;

    // truncation toward zero (matches astype(int32)), then clip, then weights
    int x0 = (int)px, y0 = (int)py;
    int x1 = x0 + 1, y1 = y0 + 1;
    const int x0c = min(max(x0, 0), kImgW - 1);
    const int x1c = min(max(x1, 0), kImgW - 1);
    const int y0c = min(max(y0, 0), kImgH - 1);
    const int y1c = min(max(y1, 0), kImgH - 1);

    const float x0f = (float)x0c, x1f = (float)x1c;
    const float y0f = (float)y0c, y1f = (float)y1c;
    const float wa = (x1f - px) * (y1f - py);
    const float wb = (x1f - px) * (py - y0f);
    const float wc = (px - x0f) * (y1f - py);
    const float wd = (px - x0f) * (py - y0f);

    // gather 4 corners, 8 channels per lane (2 x b128 each) -> stays L2-resident
    const size_t ibase = (size_t)b * (size_t)(kImgH * kImgW * kC) + (size_t)(hi * 8);
    const float* pa = img + ibase + (size_t)(y0c * kImgW + x0c) * kC;
    const float* pb = img + ibase + (size_t)(y1c * kImgW + x0c) * kC;
    const float* pc = img + ibase + (size_t)(y0c * kImgW + x1c) * kC;
    const float* pd = img + ibase + (size_t)(y1c * kImgW + x1c) * kC;

    const v4f va0 = *(const v4f*)(pa),     va1 = *(const v4f*)(pa + 4);
    const v4f vb0 = *(const v4f*)(pb),     vb1 = *(const v4f*)(pb + 4);
    const v4f vc0 = *(const v4f*)(pc),     vc1 = *(const v4f*)(pc + 4);
    const v4f vd0 = *(const v4f*)(pd),     vd1 = *(const v4f*)(pd + 4);

    const v4f o0 = wa * va0 + wb * vb0 + wc * vc0 + wd * vd0;
    const v4f o1 = wa * va1 + wb * vb1 + wc * vc1 + wd * vd1;

    // write-once output: non-temporal so it doesn't evict the image from L2
    float* po = out + (size_t)(b * (kOutH * kOutW) + oy * kOutW + ox) * kC + hi * 8;
    __builtin_nontemporal_store(o0, (v4f*)po);
    __builtin_nontemporal_store(o1, (v4f*)(po + 4));
}

extern "C" void kernel_launch(void* const* d_in, const int* in_sizes, int n_in,
                              void* d_out, int out_size, void* d_ws, size_t ws_size,
                              hipStream_t stream) {
    const float* img   = (const float*)d_in[0];   // [32,256,256,16] f32
    const float* theta = (const float*)d_in[1];   // [32,6] f32
    float* out = (float*)d_out;                   // [32,224,224,16] f32

    const int total_pixels = kBatch * kOutH * kOutW;   // 1,605,632
    const int tiles  = total_pixels / kPixPerWave;     // 100,352 (exact)
    const int blocks = tiles / kWavesPerBlock;         // 12,544  (exact)
    st_bilinear_wmma<<<blocks, 256, 0, stream>>>(img, theta, out);
}